// delta_layer_30545807409263
// MI455X (gfx1250) — compile-verified
//
#include <hip/hip_runtime.h>

typedef __attribute__((ext_vector_type(2))) float v2f;
typedef __attribute__((ext_vector_type(8))) float v8f;
typedef __attribute__((ext_vector_type(4))) float f4v;

#define TT 8   // t-tiles (of 16 rows) swept per wave

// ---------------------------------------------------------------------------
// Delta operator as tiled WMMA GEMM:
//   dst[t, d] = sum_{j in {-2,-1,1,2}} coeff(j) * src[clamp(t+j), d]
// One wave computes TT consecutive 16(t) x 16(d) output tiles:
//   D(16x16) = sum_{c=0..4} A_c(16x4) x B_c(4x16)     (V_WMMA_F32_16X16X4_F32)
// A is the fixed banded coefficient matrix (built once per wave). Edge
// replication is handled by clamped addressing in a chunk-level, wave-uniform
// (SGPR, s_cbranch) slow path taken only by the first/last chunk per row.
// All decomposition is forced scalar via readfirstlane; hot-path memory ops
// are saddr + one per-lane voffset + 24-bit immediate.
// SS = src row stride, NTSTORE = use non-temporal stores (slice not re-read).
// ---------------------------------------------------------------------------
template <int SS, bool NTSTORE>
__global__ void __launch_bounds__(256)
delta_wmma_kernel(const float* __restrict__ src, float* __restrict__ dst)
{
    constexpr int DS     = 768;   // dst row stride (3*D)
    constexpr int T      = 4096;
    constexpr int TILESD = 16;    // 256/16
    constexpr int TCH    = 32;    // (4096/16)/TT

    const int lane = threadIdx.x & 31;
    // Force wave-uniform decomposition into SGPRs.
    const int wid = __builtin_amdgcn_readfirstlane(blockIdx.x * 8 +
                                                   (threadIdx.x >> 5));
    const int dTile = wid & (TILESD - 1);
    const int tc    = (wid >> 4) & (TCH - 1);
    const int b     = wid >> 9;

    const int n  = lane & 15;   // tile column (channel); also M for A rows
    const int hi = lane >> 4;   // lane-half selector (K-half / M-half)
    const int colOff = dTile * 16 + n;
    const int tBase  = tc * (TT * 16);          // scalar

    // Scalar batch bases (SGPR pair) + 32-bit per-lane element offsets.
    const float* __restrict__ srcB = src + (size_t)(b * (T * SS));
    float*       __restrict__ dstB = dst + (size_t)(b * (T * DS));
    const int loadOff  = (tBase - 2 + 2 * hi) * SS + colOff;  // fast path only
    const int storeOff = (tBase + 8 * hi) * DS + colOff;

    // Fixed banded coefficient matrix: A[c][v] = coeff(kg - 2 - m),
    // kg = 4c + 2*hi + v, m = n. Built once per wave (branch-free selects).
    v2f A[5];
#pragma unroll
    for (int c = 0; c < 5; ++c) {
#pragma unroll
        for (int v = 0; v < 2; ++v) {
            const int off = 4 * c + 2 * hi + v - 2 - n;
            A[c][v] = (off ==  1 ?  0.50f : 0.0f)
                    + (off == -1 ? -0.50f : 0.0f)
                    + (off ==  2 ?  0.25f : 0.0f)
                    + (off == -2 ? -0.25f : 0.0f);
        }
    }

    if (tBase >= 2 && tBase + TT * 16 + 1 <= T - 1) {
        // Interior chunk (30 of 32): straight-line, immediate-offset hot loop.
#pragma unroll
        for (int tt = 0; tt < TT; ++tt) {
            v8f acc = {};
#pragma unroll
            for (int c = 0; c < 5; ++c) {
                v2f bm;
                bm[0] = srcB[loadOff + (tt * 16 + 4 * c + 0) * SS];
                bm[1] = srcB[loadOff + (tt * 16 + 4 * c + 1) * SS];
                acc = __builtin_amdgcn_wmma_f32_16x16x4_f32(
                    false, A[c], false, bm, (short)0, acc, false, false);
            }
#pragma unroll
            for (int v = 0; v < 8; ++v) {
                float* p = dstB + storeOff + (tt * 16 + v) * DS;
                if (NTSTORE) __builtin_nontemporal_store(acc[v], p);
                else         *p = acc[v];
            }
        }
    } else {
        // Boundary chunk: clamped rows (edge replication), 32-bit index math.
#pragma unroll
        for (int tt = 0; tt < TT; ++tt) {
            const int t0 = tBase + tt * 16;
            v8f acc = {};
#pragma unroll
            for (int c = 0; c < 5; ++c) {
                v2f bm;
#pragma unroll
                for (int v = 0; v < 2; ++v) {
                    int tin = t0 - 2 + 4 * c + 2 * hi + v;
                    tin = tin < 0 ? 0 : (tin > T - 1 ? T - 1 : tin);
                    bm[v] = srcB[tin * SS + colOff];
                }
                acc = __builtin_amdgcn_wmma_f32_16x16x4_f32(
                    false, A[c], false, bm, (short)0, acc, false, false);
            }
#pragma unroll
            for (int v = 0; v < 8; ++v) {
                float* p = dstB + storeOff + (tt * 16 + v) * DS;
                if (NTSTORE) __builtin_nontemporal_store(acc[v], p);
                else         *p = acc[v];
            }
        }
    }
}

// ---------------------------------------------------------------------------
// Vectorized copy of x into out[:, :, 0:D] (float4, coalesced, NT stores:
// this slice is never re-read, keep L2 free for the delta slice).
// ---------------------------------------------------------------------------
__global__ void __launch_bounds__(256)
copy_rows_kernel(const f4v* __restrict__ src, float* __restrict__ dst)
{
    const long long idx = (long long)blockIdx.x * 256 + threadIdx.x;
    const long long row = idx >> 6;          // D/4 = 64 float4 per row
    const int col = (int)(idx & 63);
    const f4v val = src[idx];
    __builtin_nontemporal_store(val, (f4v*)(dst + row * 768 + col * 4));
}

extern "C" void kernel_launch(void* const* d_in, const int* in_sizes, int n_in,
                              void* d_out, int out_size, void* d_ws, size_t ws_size,
                              hipStream_t stream) {
    (void)n_in; (void)out_size; (void)d_ws; (void)ws_size; (void)in_sizes;

    const float* x = (const float*)d_in[0];
    float* out = (float*)d_out;

    const int B = 32, T = 4096, D = 256;

    // 1) out[:, :, 0:D] = x
    {
        const long long total = (long long)B * T * (D / 4);  // float4 elements
        copy_rows_kernel<<<(int)(total / 256), 256, 0, stream>>>(
            (const f4v*)x, out);
    }

    const int waves   = B * (D / 16) * ((T / 16) / TT);  // 16384
    const int blocksW = waves / 8;                       // 8 waves per block

    // 2) out[:, :, D:2D] = delta(x)  — RT stores (re-read by pass 3 via L2).
    delta_wmma_kernel<256, false><<<blocksW, 256, 0, stream>>>(x, out + D);

    // 3) out[:, :, 2D:3D] = delta(delta(x)) — reads the delta slice (128 MiB,
    //    fits in the 192 MB L2); NT stores, matching the reference's
    //    replicate-of-delta edge semantics exactly.
    delta_wmma_kernel<768, true><<<blocksW, 256, 0, stream>>>(out + D,
                                                              out + 2 * D);
}